// LipschitzNorm_57174604644688
// MI455X (gfx1250) — compile-verified
//
#include <hip/hip_runtime.h>

#define H  8
#define C  16
#define EPS 1e-12f
#define ATT_NORM 4.0f

typedef __attribute__((ext_vector_type(2))) float v2f;
typedef __attribute__((ext_vector_type(8))) float v8f;

// ---------------------------------------------------------------------------
// Kernel 0: zero the per-(node,head) segment-max buffer (uint bits of 0.0f)
// and compute norm_att[h] = 4*sqrt(sum_c att_l^2 + att_r^2). N is read on
// device (it arrives as a device scalar), so we grid-stride.
// ---------------------------------------------------------------------------
__global__ __launch_bounds__(256) void ln_init_ws(
    unsigned int* __restrict__ seg, float* __restrict__ att_norm,
    const float* __restrict__ att_l, const float* __restrict__ att_r,
    const int* __restrict__ num_nodes_p)
{
    const int N = *num_nodes_p;
    const long long total  = (long long)N * H;
    const long long stride = (long long)gridDim.x * blockDim.x;
    for (long long i = (long long)blockIdx.x * blockDim.x + threadIdx.x;
         i < total; i += stride)
        seg[i] = 0u;                       // bits of +0.0f (norms are >= 0)

    if (blockIdx.x == 0 && threadIdx.x < H) {
        const int h = threadIdx.x;
        float s = 0.f;
        #pragma unroll
        for (int c = 0; c < C; ++c) {
            const float l = att_l[h * C + c];
            const float r = att_r[h * C + c];
            s += l * l + r * r;
        }
        att_norm[h] = ATT_NORM * sqrtf(s);
    }
}

// ---------------------------------------------------------------------------
// Kernel 1: per wave, one 16-row tile (row = e*H + h). norm_x = X^2 * ones
// via V_WMMA_F32_16X16X4_F32 (exact fp32), accumulated over 4 K-chunks.
// A layout (16x4 f32): lanes 0-15 supply K0/K1, lanes 16-31 supply K2/K3,
// row M = lane & 15.  D: every column n holds the row sum -> lane l grabs
// acc[l&7] and a shuffle routes norm[row] to lane row (0..15), which then
// does a coalesced store of norm_x into d_out and a u32 atomic max scatter.
// ---------------------------------------------------------------------------
__global__ __launch_bounds__(256) void ln_pass1_norm_wmma(
    const float* __restrict__ x, const int* __restrict__ index,
    float* __restrict__ norm_out, unsigned int* __restrict__ seg, int EH)
{
    const int lane  = threadIdx.x & 31;
    const long long wid = (long long)blockIdx.x * (blockDim.x >> 5)
                        + (threadIdx.x >> 5);
    const int tiles = (EH + 15) >> 4;
    if (wid >= tiles) return;              // wave-uniform: EXEC stays all-1s

    const int row_base = (int)(wid << 4);
    const int m  = lane & 15;              // matrix row this lane feeds
    const int hi = lane >> 4;              // 0: K={0,1}, 1: K={2,3}
    int row = row_base + m;
    if (row >= EH) row = EH - 1;           // clamp tail loads (stores guarded)
    const float* xr = x + (size_t)row * C;

    v8f acc = {};
    v2f ones; ones[0] = 1.0f; ones[1] = 1.0f;
    #pragma unroll
    for (int chunk = 0; chunk < 4; ++chunk) {
        const float2 a2 = *(const float2*)(xr + chunk * 4 + hi * 2);
        v2f A; A[0] = a2.x * a2.x; A[1] = a2.y * a2.y;
        acc = __builtin_amdgcn_wmma_f32_16x16x4_f32(
            /*neg_a=*/false, A, /*neg_b=*/false, ones,
            /*c_mod=*/(short)0, acc, /*reuse_a=*/false, /*reuse_b=*/false);
    }

    // sel = acc[lane & 7]  (rows 0-7 live in lanes 0-15, rows 8-15 in 16-31)
    const int l7 = lane & 7;
    float sel = acc[0];
    sel = (l7 == 1) ? acc[1] : sel;
    sel = (l7 == 2) ? acc[2] : sel;
    sel = (l7 == 3) ? acc[3] : sel;
    sel = (l7 == 4) ? acc[4] : sel;
    sel = (l7 == 5) ? acc[5] : sel;
    sel = (l7 == 6) ? acc[6] : sel;
    sel = (l7 == 7) ? acc[7] : sel;
    const int src = (lane < 8) ? lane : lane + 8;   // lane l<-row l's holder
    const float norm = __shfl(sel, src, 32);

    if (lane < 16) {
        const int r = row_base + lane;
        if (r < EH) {
            norm_out[r] = norm;                     // coalesced 64B store
            const int e = r >> 3;                   // r / H  (H == 8)
            const int h = r & 7;                    // r % H
            const int node = index[e];
            atomicMax(&seg[node * H + h], __float_as_uint(norm));
        }
    }
}

// ---------------------------------------------------------------------------
// Kernel 2: gather seg max (3.2MB -> L2 resident), finish the formula.
// Reads norm_x back from d_out and overwrites it with the result.
// ---------------------------------------------------------------------------
__global__ __launch_bounds__(256) void ln_pass2_finish(
    const float* __restrict__ alpha, const int* __restrict__ index,
    const unsigned int* __restrict__ seg, const float* __restrict__ att_norm,
    float* __restrict__ out, int EH)
{
    const int i = blockIdx.x * blockDim.x + threadIdx.x;
    if (i >= EH) return;
    const float norm_x = out[i];
    const int e = i >> 3;                  // H == 8
    const int h = i & 7;
    const int node = index[e];
    const float segv = __uint_as_float(seg[node * H + h]);
    const float maxn = sqrtf(segv + norm_x);
    out[i] = alpha[i] / (att_norm[h] * maxn + EPS);
}

// ---------------------------------------------------------------------------
extern "C" void kernel_launch(void* const* d_in, const int* in_sizes, int n_in,
                              void* d_out, int out_size, void* d_ws, size_t ws_size,
                              hipStream_t stream)
{
    const float* x         = (const float*)d_in[0];   // [E,H,C]
    const float* att_l     = (const float*)d_in[1];   // [1,H,C]
    const float* att_r     = (const float*)d_in[2];   // [1,H,C]
    const float* alpha     = (const float*)d_in[3];   // [E,H,1]
    const int*   index     = (const int*)  d_in[4];   // [E]
    const int*   num_nodes = (const int*)  d_in[5];   // device scalar

    float*        out      = (float*)d_out;           // [E,H,1] == EH floats
    float*        att_norm = (float*)d_ws;            // 8 floats
    unsigned int* seg      = (unsigned int*)d_ws + 8; // N*H uints

    const int EH = out_size;                          // E * H

    ln_init_ws<<<1024, 256, 0, stream>>>(seg, att_norm, att_l, att_r, num_nodes);

    const int tiles   = (EH + 15) / 16;
    const int wpb     = 256 / 32;                     // waves per block
    const int blocks1 = (tiles + wpb - 1) / wpb;
    ln_pass1_norm_wmma<<<blocks1, 256, 0, stream>>>(x, index, out, seg, EH);

    const int blocks2 = (EH + 255) / 256;
    ln_pass2_finish<<<blocks2, 256, 0, stream>>>(alpha, index, seg, att_norm, out, EH);
}